// ScheduleNet_43465069035736
// MI455X (gfx1250) — compile-verified
//
#include <hip/hip_runtime.h>

typedef __attribute__((ext_vector_type(2))) float v2f;
typedef __attribute__((ext_vector_type(8))) float v8f;

namespace {

constexpr int H  = 32;     // hidden
constexpr int T  = 5;      // node types
constexpr int N  = 5000;   // nodes
constexpr int E  = 50000;  // edges
constexpr int NT = 256;    // feasible tasks
constexpr int NF = 64;     // node feature dim
constexpr int EF = 32;     // edge feature dim
constexpr int L  = 2;      // layers
constexpr int NID = H + T * H; // 192, node-MI input dim

// ---------------------------------------------------------------- WMMA core
__device__ __forceinline__ v8f wmma_f32_k4(v2f a, v2f b, v8f c) {
  // D = A(16x4,f32) * B(4x16,f32) + C(16x16,f32)
  return __builtin_amdgcn_wmma_f32_16x16x4_f32(false, a, false, b, (short)0, c,
                                               false, false);
}

// ---------------------------------------------------------------- utilities
__global__ void k_fill_f32(float* __restrict__ p, float v, int n) {
  int i = blockIdx.x * blockDim.x + threadIdx.x;
  if (i < n) p[i] = v;
}
__global__ void k_fill_u32(unsigned* __restrict__ p, unsigned v, int n) {
  int i = blockIdx.x * blockDim.x + threadIdx.x;
  if (i < n) p[i] = v;
}

// out[r,o] = (relu?)(b[o] + sum_k in[r,k]*w[k*O+o])   (jax x @ w + b)
__global__ void k_linear(const float* __restrict__ in, const float* __restrict__ w,
                         const float* __restrict__ b, float* __restrict__ out,
                         int rows, int K, int O, int relu) {
  int idx = blockIdx.x * blockDim.x + threadIdx.x;
  if (idx >= rows * O) return;
  int r = idx / O, o = idx - r * O;
  const float* xr = in + (size_t)r * K;
  float acc = b[o];
  for (int k = 0; k < K; ++k) acc = fmaf(xr[k], w[(size_t)k * O + o], acc);
  if (relu) acc = fmaxf(acc, 0.0f);
  out[idx] = acc;
}

// -------------------------------------------------- bilinear MI via f32 WMMA
// out[b,o] = sum_{c,i} ctx[b,c]*ei[b,i]*wg[c,i,o] + sum_i ei[b,i]*bg[i,o] + bias[o]
// Restructured as Z(M x 33*I) @ W(33*I x 32); Z built on the fly in A-frag layout.
// MTILES M-tiles per wave share every streamed B fragment (halves VMEM/WMMA).
template <int IDIM, int NOUT, int MTILES>
__global__ __launch_bounds__(128) void k_mi(
    const float* __restrict__ ein, const float* __restrict__ ctx,
    const float* __restrict__ wg0, const float* __restrict__ bg0,
    const float* __restrict__ bias0, float* __restrict__ out0,
    const float* __restrict__ wg1, const float* __restrict__ bg1,
    const float* __restrict__ bias1, float* __restrict__ out1, int rows) {
  constexpr int WAVES = 4;
  constexpr int MR = 16 * MTILES;   // rows per wave
  constexpr int EIS = IDIM + 1;     // padded stride -> conflict-free LDS
  constexpr int CTS = H + 1;
  __shared__ float s_ei[WAVES][MR * EIS];
  __shared__ float s_cx[WAVES][MR * CTS];
  const int wave = threadIdx.x >> 5;
  const int lane = threadIdx.x & 31;
  const int tile = (blockIdx.x * WAVES + wave) * MR;

  // Stage this wave's MR-row tile of ei / ctx into LDS (zero-pad tail rows).
  for (int t = lane; t < MR * IDIM; t += 32) {
    int r = t / IDIM, i = t - r * IDIM;
    int gr = tile + r;
    s_ei[wave][r * EIS + i] = (gr < rows) ? ein[(size_t)gr * IDIM + i] : 0.0f;
  }
  for (int t = lane; t < MR * H; t += 32) {
    int r = t >> 5, c = t & 31;
    int gr = tile + r;
    s_cx[wave][r * CTS + c] = (gr < rows) ? ctx[(size_t)gr * H + c] : 0.0f;
  }
  __syncthreads();

  // A-frag (16x4 f32): lane m=lane&15 holds row m, k = 2*(lane>>4) + vgpr
  // B-frag (4x16 f32): lane n=lane&15 holds col n, same k mapping
  const int m = lane & 15;
  const int koff = (lane >> 4) << 1;
  const float* sei[MTILES];
  const float* scx[MTILES];
#pragma unroll
  for (int mt = 0; mt < MTILES; ++mt) {
    sei[mt] = &s_ei[wave][(mt * 16 + m) * EIS];
    scx[mt] = &s_cx[wave][(mt * 16 + m) * CTS];
  }

  v8f acc[MTILES * 2 * NOUT] = {};

  for (int c = 0; c <= H; ++c) {  // c == H folds in the Bg term (ctx = 1)
    float cv[MTILES];
#pragma unroll
    for (int mt = 0; mt < MTILES; ++mt) cv[mt] = (c < H) ? scx[mt][c] : 1.0f;
    const float* W0 = (c < H) ? (wg0 + (size_t)c * (IDIM * H)) : bg0;
    const float* W1 = nullptr;
    if constexpr (NOUT == 2) W1 = (c < H) ? (wg1 + (size_t)c * (IDIM * H)) : bg1;
    // Prefetch next c-block of the streamed (L2-resident) weights.
    __builtin_prefetch(W0 + IDIM * H + (lane << 5), 0, 3);
    if constexpr (NOUT == 2) __builtin_prefetch(W1 + IDIM * H + (lane << 5), 0, 3);

    for (int i0 = 0; i0 < IDIM; i0 += 4) {
      const int k = i0 + koff;
      v2f a[MTILES];
#pragma unroll
      for (int mt = 0; mt < MTILES; ++mt) {
        a[mt].x = cv[mt] * sei[mt][k];
        a[mt].y = cv[mt] * sei[mt][k + 1];
      }
      const float* wp0 = W0 + (size_t)k * H + m;
      v2f b00; b00.x = wp0[0];  b00.y = wp0[H];
      v2f b01; b01.x = wp0[16]; b01.y = wp0[H + 16];
#pragma unroll
      for (int mt = 0; mt < MTILES; ++mt) {
        acc[mt * 2 * NOUT + 0] = wmma_f32_k4(a[mt], b00, acc[mt * 2 * NOUT + 0]);
        acc[mt * 2 * NOUT + 1] = wmma_f32_k4(a[mt], b01, acc[mt * 2 * NOUT + 1]);
      }
      if constexpr (NOUT == 2) {
        const float* wp1 = W1 + (size_t)k * H + m;
        v2f b10; b10.x = wp1[0];  b10.y = wp1[H];
        v2f b11; b11.x = wp1[16]; b11.y = wp1[H + 16];
#pragma unroll
        for (int mt = 0; mt < MTILES; ++mt) {
          acc[mt * 2 * NOUT + 2] = wmma_f32_k4(a[mt], b10, acc[mt * 2 * NOUT + 2]);
          acc[mt * 2 * NOUT + 3] = wmma_f32_k4(a[mt], b11, acc[mt * 2 * NOUT + 3]);
        }
      }
    }
  }

  // C/D layout: vgpr v holds row v (+8 for lanes 16..31), col = lane&15
#pragma unroll
  for (int mt = 0; mt < MTILES; ++mt) {
    const int rbase = tile + mt * 16 + ((lane >> 4) << 3);
#pragma unroll
    for (int t = 0; t < NOUT; ++t) {
      const float* bias = (t == 0) ? bias0 : bias1;
      float* out = (t == 0) ? out0 : out1;
#pragma unroll
      for (int half = 0; half < 2; ++half) {
        v8f v = acc[mt * 2 * NOUT + t * 2 + half];
        float bv = bias[half * 16 + m];
        for (int r = 0; r < 8; ++r) {
          int row = rbase + r;
          if (row < rows) out[(size_t)row * H + half * 16 + m] = v[r] + bv;
        }
      }
    }
  }
}

// ---------------------------------------------------------------- gathers
__global__ void k_edge_inputs(const int* __restrict__ src, const int* __restrict__ dst,
                              const int* __restrict__ ntype, const float* __restrict__ x,
                              const float* __restrict__ ea, const float* __restrict__ temb,
                              float* __restrict__ ei, float* __restrict__ ctx) {
  int idx = blockIdx.x * blockDim.x + threadIdx.x;
  if (idx >= E * H) return;
  int e = idx >> 5, h = idx & 31;
  int s = src[e], d = dst[e];
  ei[(size_t)e * (3 * H) + h]         = x[(size_t)d * H + h];
  ei[(size_t)e * (3 * H) + H + h]     = x[(size_t)s * H + h];
  ei[(size_t)e * (3 * H) + 2 * H + h] = ea[(size_t)e * H + h];
  ctx[(size_t)e * H + h] = temb[(size_t)ntype[s] * H + h];
}

__global__ void k_node_inputs(const int* __restrict__ ntype, const float* __restrict__ x,
                              const float* __restrict__ agg, const float* __restrict__ temb,
                              float* __restrict__ nin, float* __restrict__ ctx) {
  int idx = blockIdx.x * blockDim.x + threadIdx.x;
  if (idx >= N * NID) return;
  int n = idx / NID, j = idx - n * NID;
  nin[idx] = (j < H) ? x[(size_t)n * H + j] : agg[(size_t)n * (T * H) + (j - H)];
  if (j < H) ctx[(size_t)n * H + j] = temb[(size_t)ntype[n] * H + j];
}

// ------------------------------------------------------- segment softmax/agg
__device__ __forceinline__ unsigned fkey(float f) {  // monotonic uint key
  unsigned u = __float_as_uint(f);
  return (u & 0x80000000u) ? ~u : (u | 0x80000000u);
}
__device__ __forceinline__ float funkey(unsigned k) {
  unsigned u = (k & 0x80000000u) ? (k & 0x7fffffffu) : ~k;
  return __uint_as_float(u);
}

__global__ void k_seg_max(const int* __restrict__ src, const int* __restrict__ dst,
                          const int* __restrict__ ntype, const float* __restrict__ logits,
                          unsigned* __restrict__ mkey) {
  int e = blockIdx.x * blockDim.x + threadIdx.x;
  if (e >= E) return;
  int seg = dst[e] * T + ntype[src[e]];
  atomicMax(&mkey[seg], fkey(logits[e]));
}

__global__ void k_seg_expsum(const int* __restrict__ src, const int* __restrict__ dst,
                             const int* __restrict__ ntype, const float* __restrict__ logits,
                             const unsigned* __restrict__ mkey,
                             float* __restrict__ ex, float* __restrict__ ssum) {
  int e = blockIdx.x * blockDim.x + threadIdx.x;
  if (e >= E) return;
  int seg = dst[e] * T + ntype[src[e]];
  float mv = funkey(mkey[seg]);
  if (!(mv > -3.0e38f && mv < 3.0e38f)) mv = 0.0f;  // guard (matches isfinite)
  float v = expf(logits[e] - mv);
  ex[e] = v;
  atomicAdd(&ssum[seg], v);
}

__global__ void k_seg_agg(const int* __restrict__ src, const int* __restrict__ dst,
                          const int* __restrict__ ntype, const float* __restrict__ ex,
                          const float* __restrict__ ssum, const float* __restrict__ enew,
                          float* __restrict__ agg) {
  int idx = blockIdx.x * blockDim.x + threadIdx.x;
  if (idx >= E * H) return;
  int e = idx >> 5, h = idx & 31;
  int seg = dst[e] * T + ntype[src[e]];
  float w = ex[e] / ssum[seg];
  atomicAdd(&agg[(size_t)seg * H + h], w * enew[(size_t)e * H + h]);
}

// ---------------------------------------------------------------- head
__global__ void k_find_ee(const int* __restrict__ src, const int* __restrict__ dst,
                          const int* __restrict__ agent, const int* __restrict__ feas,
                          int* __restrict__ ee_idx) {
  int e = blockIdx.x * blockDim.x + threadIdx.x;
  if (e >= E) return;
  if (src[e] != agent[0]) return;
  int d = dst[e];
  for (int f = 0; f < NT; ++f)
    if (feas[f] == d) atomicMin(&ee_idx[f], e);  // first match == argmax(match)
}

__global__ void k_build_comb(const float* __restrict__ x, const float* __restrict__ ea,
                             const int* __restrict__ agent, const int* __restrict__ feas,
                             const int* __restrict__ ee_idx, float* __restrict__ comb) {
  int idx = blockIdx.x * blockDim.x + threadIdx.x;
  if (idx >= NT * 3 * H) return;
  int f = idx / (3 * H), j = idx - f * (3 * H);
  float v;
  if (j < H)            v = x[(size_t)agent[0] * H + j];
  else if (j < 2 * H)   v = x[(size_t)feas[f] * H + (j - H)];
  else {
    int e = ee_idx[f];
    v = (e == 0x7fffffff) ? 0.0f : ea[(size_t)e * H + (j - 2 * H)];
  }
  comb[idx] = v;
}

}  // namespace

extern "C" void kernel_launch(void* const* d_in, const int* in_sizes, int n_in,
                              void* d_out, int out_size, void* d_ws, size_t ws_size,
                              hipStream_t stream) {
  (void)in_sizes; (void)n_in; (void)out_size; (void)ws_size;
  const float* node_feat = (const float*)d_in[0];
  const int*   e_src     = (const int*)d_in[1];
  const int*   e_dst     = e_src + E;
  const float* edge_feat = (const float*)d_in[2];
  const int*   ntype     = (const int*)d_in[3];
  const int*   agent     = (const int*)d_in[4];
  const int*   feas      = (const int*)d_in[5];
  auto P = [&](int i) { return (const float*)d_in[i]; };

  // params flattened in jax pytree order (dict keys sorted, lists in order)
  const float *a0b = P(6),  *a0w = P(7),  *a1b = P(8),  *a1w = P(9);
  const float *a2b = P(10), *a2w = P(11);
  const float *ee0b = P(12), *ee0w = P(13), *ee1b = P(14), *ee1w = P(15);
  const float *ne0b = P(16), *ne0w = P(17), *ne1b = P(18), *ne1w = P(19);

  // ---- workspace carve ----
  float* wsp = (float*)d_ws;
  auto alloc = [&](size_t n) { float* p = wsp; wsp += n; return p; };
  float* x_a   = alloc((size_t)N * H);
  float* x_b   = alloc((size_t)N * H);
  float* ea_a  = alloc((size_t)E * H);
  float* ea_b  = alloc((size_t)E * H);
  float* ei_e  = alloc((size_t)E * 3 * H);
  float* ctx_e = alloc((size_t)E * H);
  float* mi_e  = alloc((size_t)E * H);
  float* mi_a  = alloc((size_t)E * H);
  float* tmp_e = alloc((size_t)E * H);
  float* logits = alloc(E);
  float* ex    = alloc(E);
  unsigned* mkey = (unsigned*)alloc(N * T);
  float* ssum  = alloc(N * T);
  float* agg   = alloc((size_t)N * T * H);
  float* nin   = alloc((size_t)N * NID);
  float* ctx_n = alloc((size_t)N * H);
  float* tn1   = alloc((size_t)N * H);
  float* tn2   = alloc((size_t)N * H);
  float* comb  = alloc((size_t)NT * 3 * H);
  float* h1    = alloc((size_t)NT * 256);
  float* h2    = alloc((size_t)NT * 128);
  int* ee_idx  = (int*)alloc(NT);

  const int B = 256;
  auto g = [](long n, int b) { return dim3((unsigned)((n + b - 1) / b)); };

  // ---- encoders ----
  k_linear<<<g((long)N * H, B), B, 0, stream>>>(node_feat, ne0w, ne0b, tn1, N, NF, H, 1);
  k_linear<<<g((long)N * H, B), B, 0, stream>>>(tn1, ne1w, ne1b, x_a, N, H, H, 0);
  k_linear<<<g((long)E * H, B), B, 0, stream>>>(edge_feat, ee0w, ee0b, tmp_e, E, EF, H, 1);
  k_linear<<<g((long)E * H, B), B, 0, stream>>>(tmp_e, ee1w, ee1b, ea_a, E, H, H, 0);

  float* x = x_a;  float* xn = x_b;
  float* ea = ea_a; float* ean = ea_b;

  for (int l = 0; l < L; ++l) {
    int base = 20 + l * 22;
    const float *attn_bg = P(base + 0), *attn_bias = P(base + 1), *attn_wg = P(base + 2);
    const float *am0b = P(base + 3), *am0w = P(base + 4);
    const float *am1b = P(base + 5), *am1w = P(base + 6);
    const float *edge_bg = P(base + 7), *edge_bias = P(base + 8), *edge_wg = P(base + 9);
    const float *em0b = P(base + 10), *em0w = P(base + 11);
    const float *em1b = P(base + 12), *em1w = P(base + 13);
    const float *node_bg = P(base + 14), *node_bias = P(base + 15), *node_wg = P(base + 16);
    const float *nm0b = P(base + 17), *nm0w = P(base + 18);
    const float *nm1b = P(base + 19), *nm1w = P(base + 20);
    const float *temb = P(base + 21);

    k_edge_inputs<<<g((long)E * H, B), B, 0, stream>>>(e_src, e_dst, ntype, x, ea, temb,
                                                       ei_e, ctx_e);
    // edge-MI + attn-MI share Z; 2 M-tiles/wave share every B fragment
    k_mi<3 * H, 2, 2><<<dim3((E + 127) / 128), 128, 0, stream>>>(
        ei_e, ctx_e, edge_wg, edge_bg, edge_bias, mi_e,
        attn_wg, attn_bg, attn_bias, mi_a, E);
    // edge mlp -> e_new
    k_linear<<<g((long)E * H, B), B, 0, stream>>>(mi_e, em0w, em0b, tmp_e, E, H, H, 1);
    k_linear<<<g((long)E * H, B), B, 0, stream>>>(tmp_e, em1w, em1b, ean, E, H, H, 0);
    // attn mlp -> logits
    k_linear<<<g((long)E * H, B), B, 0, stream>>>(mi_a, am0w, am0b, tmp_e, E, H, H, 1);
    k_linear<<<g(E, B), B, 0, stream>>>(tmp_e, am1w, am1b, logits, E, H, 1, 0);
    // type-aware segment softmax + weighted aggregation
    k_fill_u32<<<g(N * T, B), B, 0, stream>>>(mkey, 0u, N * T);
    k_fill_f32<<<g(N * T, B), B, 0, stream>>>(ssum, 0.0f, N * T);
    k_fill_f32<<<g((long)N * T * H, B), B, 0, stream>>>(agg, 0.0f, N * T * H);
    k_seg_max<<<g(E, B), B, 0, stream>>>(e_src, e_dst, ntype, logits, mkey);
    k_seg_expsum<<<g(E, B), B, 0, stream>>>(e_src, e_dst, ntype, logits, mkey, ex, ssum);
    k_seg_agg<<<g((long)E * H, B), B, 0, stream>>>(e_src, e_dst, ntype, ex, ssum, ean, agg);
    // node update (MTILES=1: keep 79 blocks of parallelism for only ~2 GFLOP)
    k_node_inputs<<<g((long)N * NID, B), B, 0, stream>>>(ntype, x, agg, temb, nin, ctx_n);
    k_mi<NID, 1, 1><<<dim3((N + 63) / 64), 128, 0, stream>>>(
        nin, ctx_n, node_wg, node_bg, node_bias, tn1,
        nullptr, nullptr, nullptr, nullptr, N);
    k_linear<<<g((long)N * H, B), B, 0, stream>>>(tn1, nm0w, nm0b, tn2, N, H, H, 1);
    k_linear<<<g((long)N * H, B), B, 0, stream>>>(tn2, nm1w, nm1b, xn, N, H, H, 0);
    // ping-pong
    float* ts = x;  x = xn;  xn = ts;
    ts = ea;  ea = ean;  ean = ts;
  }

  // ---- actor head ----
  k_fill_u32<<<g(NT, B), B, 0, stream>>>((unsigned*)ee_idx, 0x7fffffffu, NT);
  k_find_ee<<<g(E, B), B, 0, stream>>>(e_src, e_dst, agent, feas, ee_idx);
  k_build_comb<<<g(NT * 3 * H, B), B, 0, stream>>>(x, ea, agent, feas, ee_idx, comb);
  k_linear<<<g((long)NT * 256, B), B, 0, stream>>>(comb, a0w, a0b, h1, NT, 3 * H, 256, 1);
  k_linear<<<g((long)NT * 128, B), B, 0, stream>>>(h1, a1w, a1b, h2, NT, 256, 128, 1);
  k_linear<<<g(NT, B), B, 0, stream>>>(h2, a2w, a2b, (float*)d_out, NT, 128, 1, 0);
}